// DFTB_34935263986162
// MI455X (gfx1250) — compile-verified
//
#include <hip/hip_runtime.h>
#include <math.h>

typedef __attribute__((ext_vector_type(2))) float v2f;
typedef __attribute__((ext_vector_type(8))) float v8f;

#define TS 96                      // padded/strided K dimension (>= 84, mult of 4; *4B is 8B-aligned per row)
#define BOHR_F 1.8897261258369282f
#define CUTOFF_F 6.0f
#define NPOLY 12

// ---------------------------------------------------------------------------
// Kernel 1: trig table T[n][s], n in [0,64): n<32 -> cos row for k=n,
// n>=32 -> sin row for k=n-32.  ang = 2*pi*0.2*dot(kp[k], shift[s])
// (inv(diag(5)) = diag(0.2)).  Zero-padded for s >= S.
// ---------------------------------------------------------------------------
__global__ __launch_bounds__(256) void dftb_trig(const float* __restrict__ kp,
                                                 const float* __restrict__ shifts,
                                                 float* __restrict__ T,
                                                 int S, int nk) {
  int idx = blockIdx.x * blockDim.x + threadIdx.x;
  if (idx >= 64 * TS) return;
  int n = idx / TS;
  int s = idx - n * TS;
  int k = n & 31;
  float val = 0.0f;
  if (s < S && k < nk) {
    float ang = 6.28318530717958647f * 0.2f *
                (kp[3 * k + 0] * shifts[3 * s + 0] +
                 kp[3 * k + 1] * shifts[3 * s + 1] +
                 kp[3 * k + 2] * shifts[3 * s + 2]);
    val = (n < 32) ? cosf(ang) : sinf(ang);
  }
  T[idx] = val;
}

// ---------------------------------------------------------------------------
// Kernel 2: per wave, one (i, j0) tile: C[64 x 16] = A[64 x K] * B[K x 16]
// A = trig table (from LDS), B = V values computed on the fly.
// WMMA f32 16x16x4; 4 M-tiles accumulated per wave.
// ---------------------------------------------------------------------------
__device__ __forceinline__ float evalV(float dx, float dy, float dz,
                                       const float c[NPOLY]) {
  float dr2 = dx * dx + dy * dy + dz * dz;
  float dr = sqrtf(dr2);
  float x = dr * BOHR_F;
  float y = c[NPOLY - 1];
#pragma unroll
  for (int p = NPOLY - 2; p >= 0; --p) y = y * x + c[p];
  return (dr > 0.1f && dr <= CUTOFF_F) ? y : 0.0f;
}

__global__ __launch_bounds__(256) void dftb_main(
    const float* __restrict__ pos, const int* __restrict__ species,
    const float* __restrict__ pm, const float* __restrict__ onsite,
    const float* __restrict__ Tg, const float* __restrict__ shifts,
    float* __restrict__ out, int na, int S, int nsteps, int nsp, int tiles) {
  __shared__ float Tlds[64 * TS];
  __shared__ float4 Shlds[TS];

  int tid = threadIdx.x;
  for (int idx = tid; idx < 64 * TS; idx += 256) Tlds[idx] = Tg[idx];
  for (int s = tid; s < TS; s += 256) {
    float4 v;
    if (s < S) {
      v.x = shifts[3 * s + 0];
      v.y = shifts[3 * s + 1];
      v.z = shifts[3 * s + 2];
      v.w = 0.0f;
    } else {
      v.x = 1.0e6f;  // forces dr > cutoff -> V = 0 for padded K (trig is 0 too)
      v.y = 0.0f; v.z = 0.0f; v.w = 0.0f;
    }
    Shlds[s] = v;
  }
  __syncthreads();

  int wave = tid >> 5;
  int lane = tid & 31;
  int tile = blockIdx.x * 8 + wave;
  if (tile >= tiles) return;

  int jt = na >> 4;          // 16-wide j tiles per atom row i
  int i  = tile / jt;
  int j0 = (tile - i * jt) << 4;
  int jn = lane & 15;        // N column within tile (also A-row selector)
  int h  = lane >> 4;        // lane half: selects K offset {0,2} per ISA layout
  int j  = j0 + jn;

  float pix = pos[3 * i + 0], piy = pos[3 * i + 1], piz = pos[3 * i + 2];
  float d0x = pos[3 * j + 0] - pix;
  float d0y = pos[3 * j + 1] - piy;
  float d0z = pos[3 * j + 2] - piz;

  int si = species[i];
  int sj = species[j];
  const float* cb = pm + ((size_t)si * (size_t)nsp + (size_t)sj) * NPOLY;
  float c[NPOLY];
#pragma unroll
  for (int p = 0; p < NPOLY; ++p) c[p] = cb[p];
  float ons = onsite[si];

  v8f acc0 = {}, acc1 = {}, acc2 = {}, acc3 = {};

  for (int step = 0; step < nsteps; ++step) {
    int s0 = 4 * step + 2 * h;        // this lane's K pair: s0, s0+1
    float4 sh0 = Shlds[s0];
    float4 sh1 = Shlds[s0 + 1];

    float V0 = evalV(d0x + sh0.x, d0y + sh0.y, d0z + sh0.z, c);
    float V1 = evalV(d0x + sh1.x, d0y + sh1.y, d0z + sh1.z, c);
    v2f b;
    b.x = V0;  // B[K=2h  ][N=jn]
    b.y = V1;  // B[K=2h+1][N=jn]

    const float* trow = &Tlds[(size_t)jn * TS + s0];
    v2f a0 = *(const v2f*)(trow + 0 * 16 * TS);   // A rows  0..15
    v2f a1 = *(const v2f*)(trow + 1 * 16 * TS);   // A rows 16..31
    v2f a2 = *(const v2f*)(trow + 2 * 16 * TS);   // A rows 32..47
    v2f a3 = *(const v2f*)(trow + 3 * 16 * TS);   // A rows 48..63

    acc0 = __builtin_amdgcn_wmma_f32_16x16x4_f32(false, a0, false, b, (short)0, acc0, false, false);
    acc1 = __builtin_amdgcn_wmma_f32_16x16x4_f32(false, a1, false, b, (short)0, acc1, false, false);
    acc2 = __builtin_amdgcn_wmma_f32_16x16x4_f32(false, a2, false, b, (short)0, acc2, false, false);
    acc3 = __builtin_amdgcn_wmma_f32_16x16x4_f32(false, a3, false, b, (short)0, acc3, false, false);
  }

  // C element (M=n, N=jn): lanes 0-15 hold rows r, lanes 16-31 rows r+8.
  size_t plane = (size_t)na * (size_t)na;
  float* obase = out + (size_t)i * (size_t)na + (size_t)j;
  float add = (j == i) ? ons : 0.0f;  // onsite only on Hr (rows 0..31)
#pragma unroll
  for (int r = 0; r < 8; ++r) {
    int nr = r + 8 * h;
    obase[(size_t)(nr + 0)  * plane] = acc0[r] + add;  // Hr, k = nr
    obase[(size_t)(nr + 16) * plane] = acc1[r] + add;  // Hr, k = nr+16
    obase[(size_t)(nr + 32) * plane] = acc2[r];        // Hi, k = nr
    obase[(size_t)(nr + 48) * plane] = acc3[r];        // Hi, k = nr+16
  }
}

// ---------------------------------------------------------------------------
extern "C" void kernel_launch(void* const* d_in, const int* in_sizes, int n_in,
                              void* d_out, int out_size, void* d_ws,
                              size_t ws_size, hipStream_t stream) {
  const float* pos     = (const float*)d_in[0];
  const int*   species = (const int*)d_in[1];
  const float* kp      = (const float*)d_in[2];
  const float* pm      = (const float*)d_in[3];
  const float* onsite  = (const float*)d_in[4];
  const float* shifts  = (const float*)d_in[5];

  int na  = in_sizes[0] / 3;   // 512
  int nk  = in_sizes[2] / 3;   // 32
  int nsp = in_sizes[4];       // 4
  int S   = in_sizes[5] / 3;   // 81
  int nsteps = (S + 3) / 4;    // 21 K-steps of 4

  float* T = (float*)d_ws;     // 64*TS floats = 24 KB trig table

  dftb_trig<<<(64 * TS + 255) / 256, 256, 0, stream>>>(kp, shifts, T, S, nk);

  int tiles = na * (na >> 4);  // 16384 waves of work
  dftb_main<<<(tiles + 7) / 8, 256, 0, stream>>>(
      pos, species, pm, onsite, T, shifts, (float*)d_out, na, S, nsteps, nsp,
      tiles);
}